// SurfaceDistanceLoss_84945863180901
// MI455X (gfx1250) — compile-verified
//
#include <hip/hip_runtime.h>
#include <hip/hip_bf16.h>
#include <math.h>

typedef float v2f __attribute__((ext_vector_type(2)));
typedef float v8f __attribute__((ext_vector_type(8)));

#define K_NN      8
#define S_PER_F   128
#define FI        512
#define FS        128
#define NPTS      (FI * S_PER_F)   /* 65536 input samples  */
#define NQRS      (FS * S_PER_F)   /* 16384 query samples  */
#define NCHUNK    (NPTS / 16)      /* 4096 point tiles     */
#define BIGF      3.0e38f

// ---------------------------------------------------------------------------
// Forward loss: barycenter cdist^2, min over input faces, dot with probs.
// One block of 512 threads; everything fits in LDS.
// ---------------------------------------------------------------------------
__global__ void fwd_kernel(const float* __restrict__ iv, const int* __restrict__ ifc,
                           const float* __restrict__ sv, const int* __restrict__ sfc,
                           const float* __restrict__ probs, float* __restrict__ fwd_out)
{
    __shared__ float bci[FI * 3];
    __shared__ float bcs[FS * 3];
    __shared__ float red[FS];
    const int t = threadIdx.x;
    if (t < FI) {
        int a = ifc[t * 3 + 0], b = ifc[t * 3 + 1], c = ifc[t * 3 + 2];
        bci[t * 3 + 0] = (iv[a * 3 + 0] + iv[b * 3 + 0] + iv[c * 3 + 0]) * (1.0f / 3.0f);
        bci[t * 3 + 1] = (iv[a * 3 + 1] + iv[b * 3 + 1] + iv[c * 3 + 1]) * (1.0f / 3.0f);
        bci[t * 3 + 2] = (iv[a * 3 + 2] + iv[b * 3 + 2] + iv[c * 3 + 2]) * (1.0f / 3.0f);
    }
    if (t < FS) {
        int a = sfc[t * 3 + 0], b = sfc[t * 3 + 1], c = sfc[t * 3 + 2];
        bcs[t * 3 + 0] = (sv[a * 3 + 0] + sv[b * 3 + 0] + sv[c * 3 + 0]) * (1.0f / 3.0f);
        bcs[t * 3 + 1] = (sv[a * 3 + 1] + sv[b * 3 + 1] + sv[c * 3 + 1]) * (1.0f / 3.0f);
        bcs[t * 3 + 2] = (sv[a * 3 + 2] + sv[b * 3 + 2] + sv[c * 3 + 2]) * (1.0f / 3.0f);
    }
    __syncthreads();
    if (t < FS) {
        float qx = bcs[t * 3 + 0], qy = bcs[t * 3 + 1], qz = bcs[t * 3 + 2];
        float best = BIGF;
        for (int j = 0; j < FI; ++j) {
            float dx = bci[j * 3 + 0] - qx;
            float dy = bci[j * 3 + 1] - qy;
            float dz = bci[j * 3 + 2] - qz;
            best = fminf(best, dx * dx + dy * dy + dz * dz);
        }
        red[t] = best * probs[t];
    }
    __syncthreads();
    for (int off = FS / 2; off > 0; off >>= 1) {
        if (t < off) red[t] += red[t + off];
        __syncthreads();
    }
    if (t == 0) fwd_out[0] = red[0];
}

// ---------------------------------------------------------------------------
// Surface sampling -> WMMA-ready float4 layout.
// mode 0 (points):  (-2x, -2y, -2z, |p|^2)   -> A matrix rows
// mode 1 (queries): ( x,   y,   z,  1    )   -> B matrix columns
// The two contiguous halves of each float4 are exactly the two K-pairs the
// two half-waves need, so the kNN kernel can load v2f directly (no cndmask).
// ---------------------------------------------------------------------------
__global__ void sample_kernel(const float* __restrict__ verts, const int* __restrict__ faces,
                              const float* __restrict__ s1r, const float* __restrict__ r2r,
                              float4* __restrict__ out, int total, int mode)
{
    const int i = blockIdx.x * blockDim.x + threadIdx.x;
    if (i >= total) return;
    const int f = i >> 7;  // /S_PER_F
    const int a = faces[f * 3 + 0], b = faces[f * 3 + 1], c = faces[f * 3 + 2];
    const float s1 = sqrtf(s1r[i]);
    const float r2 = r2r[i];
    const float wa = 1.0f - s1, wb = s1 * (1.0f - r2), wc = s1 * r2;
    const float px = wa * verts[a * 3 + 0] + wb * verts[b * 3 + 0] + wc * verts[c * 3 + 0];
    const float py = wa * verts[a * 3 + 1] + wb * verts[b * 3 + 1] + wc * verts[c * 3 + 1];
    const float pz = wa * verts[a * 3 + 2] + wb * verts[b * 3 + 2] + wc * verts[c * 3 + 2];
    if (mode == 0)
        out[i] = make_float4(-2.0f * px, -2.0f * py, -2.0f * pz, px * px + py * py + pz * pz);
    else
        out[i] = make_float4(px, py, pz, 1.0f);
}

// ---------------------------------------------------------------------------
// kNN(8) via V_WMMA_F32_16X16X4_F32.
// One wave owns 16 queries (B matrix, N dim), streams all points 16 at a
// time as A matrix (M dim).  C[m][n] = |p_m|^2 - 2 p_m . q_n ; the per-query
// |q|^2 is monotone and added after selection.  In the 16x16 f32 C layout
// lane l holds 8 rows (points) of column (query) l&15, so top-8 selection is
// a per-lane register min/max chain (compiler fuses it into dual-issued
// v_maxmin/v_minmax); lane q and lane q+16 cover the two point half-tiles
// and merge once at the end via ds_bpermute.
// Hot loop per chunk: 1x global_load_b64 + 1x v_wmma + selection chain.
// ---------------------------------------------------------------------------
__global__ __launch_bounds__(256) void knn_kernel(const float4* __restrict__ pts,
                                                  const float4* __restrict__ qrs,
                                                  float* __restrict__ per_query)
{
    const int lane  = threadIdx.x & 31;
    const int wave  = (blockIdx.x * blockDim.x + threadIdx.x) >> 5;
    const int qbase = wave * 16;
    if (qbase >= NQRS) return;            // wave-uniform (EXEC stays all-1s)
    const int n  = lane & 15;
    const int hi = lane >> 4;

    // B matrix K-pair: lanes 0-15 get (qx,qy); lanes 16-31 get (qz,1).
    const v2f* __restrict__ qrs2 = (const v2f*)qrs;
    const v2f bmat = qrs2[(qbase + n) * 2 + hi];
    const float4 q = qrs[qbase + n];      // once, for |q|^2
    const float q2 = q.x * q.x + q.y * q.y + q.z * q.z;

    float s[K_NN];
#pragma unroll
    for (int i = 0; i < K_NN; ++i) s[i] = BIGF;

    const v2f* __restrict__ pts2 = (const v2f*)pts;
    for (int ch = 0; ch < NCHUNK; ++ch) {
        // A matrix K-pair: direct 8-byte load, no lane select needed.
        const v2f amat = pts2[(ch * 16 + n) * 2 + hi];
        v8f cacc = {};
        cacc = __builtin_amdgcn_wmma_f32_16x16x4_f32(
            false, amat, false, bmat, (short)0, cacc, false, false);
#pragma unroll
        for (int i = 0; i < 8; ++i) {
            float v = cacc[i];
#pragma unroll
            for (int j = 0; j < K_NN; ++j) {   // branchless sorted-insert
                float lo = fminf(v, s[j]);
                v = fmaxf(v, s[j]);
                s[j] = lo;
            }
        }
    }

    // merge the partner half-wave's top-8 (points 8..15 vs 0..7 of each tile)
    const int paddr = ((lane ^ 16) << 2);
#pragma unroll
    for (int i = 0; i < K_NN; ++i) {
        float v = __int_as_float(__builtin_amdgcn_ds_bpermute(paddr, __float_as_int(s[i])));
#pragma unroll
        for (int j = 0; j < K_NN; ++j) {
            float lo = fminf(v, s[j]);
            v = fmaxf(v, s[j]);
            s[j] = lo;
        }
    }

    if (hi == 0) {
        float acc = 0.0f;
#pragma unroll
        for (int i = 0; i < K_NN; ++i) acc += sqrtf(fmaxf(s[i] + q2, 0.0f));
        per_query[qbase + n] = acc;
    }
}

// ---------------------------------------------------------------------------
// Per-face sum, max-normalize, scale, add forward scalar.
// ---------------------------------------------------------------------------
__global__ void finalize_kernel(const float* __restrict__ per_query,
                                const float* __restrict__ probs,
                                const float* __restrict__ fwd,
                                float* __restrict__ out)
{
    __shared__ float sm[FS];
    const int f = threadIdx.x;            // 128 threads
    float s = 0.0f;
    for (int j = 0; j < S_PER_F; ++j) s += per_query[f * S_PER_F + j];
    sm[f] = s;
    __syncthreads();
    for (int off = FS / 2; off > 0; off >>= 1) {
        if (f < off) sm[f] = fmaxf(sm[f], sm[f + off]);
        __syncthreads();
    }
    const float mx = sm[0];
    out[f] = fwd[0] + (s / mx) * 0.1f * probs[f];
}

// ---------------------------------------------------------------------------
extern "C" void kernel_launch(void* const* d_in, const int* in_sizes, int n_in,
                              void* d_out, int out_size, void* d_ws, size_t ws_size,
                              hipStream_t stream)
{
    (void)in_sizes; (void)n_in; (void)out_size; (void)ws_size;

    const float* iv   = (const float*)d_in[0];
    const int*   ifc  = (const int*)  d_in[1];
    const float* sv   = (const float*)d_in[2];
    const int*   sfc  = (const int*)  d_in[3];
    const float* prob = (const float*)d_in[4];
    const float* s1i  = (const float*)d_in[5];
    const float* r2i  = (const float*)d_in[6];
    const float* s1s  = (const float*)d_in[7];
    const float* r2s  = (const float*)d_in[8];

    char*   ws   = (char*)d_ws;
    float*  fwd  = (float*)ws;                                                // 16 B
    float4* inp  = (float4*)(ws + 16);                                        // 1 MB
    float4* simp = (float4*)(ws + 16 + (size_t)NPTS * 16);                    // 256 KB
    float*  perq = (float*)(ws + 16 + (size_t)NPTS * 16 + (size_t)NQRS * 16); // 64 KB

    fwd_kernel<<<1, 512, 0, stream>>>(iv, ifc, sv, sfc, prob, fwd);
    sample_kernel<<<NPTS / 256, 256, 0, stream>>>(iv, ifc, s1i, r2i, inp, NPTS, 0);
    sample_kernel<<<NQRS / 256, 256, 0, stream>>>(sv, sfc, s1s, r2s, simp, NQRS, 1);
    knn_kernel<<<(NQRS / 16) * 32 / 256, 256, 0, stream>>>(inp, simp, perq);
    finalize_kernel<<<1, FS, 0, stream>>>(perq, prob, fwd, (float*)d_out);
}